// Net_61229053771812
// MI455X (gfx1250) — compile-verified
//
#include <hip/hip_runtime.h>
#include <hip/hip_bf16.h>
#include <stdint.h>

typedef __attribute__((ext_vector_type(2))) float v2f;
typedef __attribute__((ext_vector_type(8))) float v8f;

#define N_NODES 89250
#define IN_F    500
#define HID     256
#define N_CLS   7

// ------------------------------------------------ CDNA5 async global->LDS
// GLOBAL_LOAD_ASYNC_TO_LDS_B32 (GV mode): each lane copies 4 bytes from the global address in vaddr to the LDS address in vdst.
// Tracked by ASYNCcnt; consumed after s_wait_asynccnt 0 + workgroup barrier.
__device__ __forceinline__ void async_copy_b32(unsigned lds_off, const void* gaddr) {
    asm volatile("global_load_async_to_lds_b32 %0, %1, off"
                 :: "v"(lds_off), "v"(gaddr) : "memory");
}
__device__ __forceinline__ void wait_async0() {
    asm volatile("s_wait_asynccnt 0" ::: "memory");
}
__device__ __forceinline__ unsigned lds_off(const void* p) {
    return (unsigned)(uintptr_t)p;   // generic LDS ptr: low 32 bits = LDS offset
}

// ---------------------------------------------------------------- utilities
__global__ void zero_int_kernel(int* __restrict__ p, int n) {
    int i = blockIdx.x * blockDim.x + threadIdx.x;
    if (i < n) p[i] = 0;
}

__global__ void csr_count_kernel(const int* __restrict__ dst, int* __restrict__ cnt, int E) {
    int e = blockIdx.x * blockDim.x + threadIdx.x;
    if (e < E) atomicAdd(&cnt[dst[e]], 1);
}

// single-block exclusive scan: ptr[0]=0, ptr[i+1]=sum(cnt[0..i])
__global__ void csr_scan_kernel(const int* __restrict__ cnt, int* __restrict__ ptr, int n) {
    __shared__ int sdata[1024];
    __shared__ int carry;
    if (threadIdx.x == 0) { carry = 0; ptr[0] = 0; }
    __syncthreads();
    for (int base = 0; base < n; base += 1024) {
        int i = base + (int)threadIdx.x;
        int v = (i < n) ? cnt[i] : 0;
        sdata[threadIdx.x] = v;
        __syncthreads();
        for (int off = 1; off < 1024; off <<= 1) {
            int t = (threadIdx.x >= (unsigned)off) ? sdata[threadIdx.x - off] : 0;
            __syncthreads();
            sdata[threadIdx.x] += t;
            __syncthreads();
        }
        if (i < n) ptr[i + 1] = carry + sdata[threadIdx.x];
        __syncthreads();
        if (threadIdx.x == 0) carry += sdata[1023];
        __syncthreads();
    }
}

__global__ void csr_fill_kernel(const int* __restrict__ dst, const int* __restrict__ ptr,
                                int* __restrict__ fill, int* __restrict__ eid, int E) {
    int e = blockIdx.x * blockDim.x + threadIdx.x;
    if (e < E) {
        int d = dst[e];
        int pos = ptr[d] + atomicAdd(&fill[d], 1);
        eid[pos] = e;
    }
}

// ------------------------------------------------------- scatter-mean (CSR)
template <int F>
__global__ void aggregate_kernel(const float* __restrict__ xin,
                                 const int* __restrict__ ptr,
                                 const int* __restrict__ eid,
                                 const int* __restrict__ src,
                                 const float* __restrict__ ew,
                                 float* __restrict__ agg, int n) {
    int node = blockIdx.x * (blockDim.x >> 5) + (threadIdx.x >> 5);
    int lane = threadIdx.x & 31;
    if (node >= n) return;
    constexpr int R = (F + 31) / 32;
    float acc[R];
#pragma unroll
    for (int r = 0; r < R; r++) acc[r] = 0.0f;
    int beg = ptr[node], end = ptr[node + 1];
    for (int p = beg; p < end; p++) {
        int e = eid[p];
        float w = ew[e];
        const float* xr = xin + (long)src[e] * F;
#pragma unroll
        for (int r = 0; r < R; r++) {
            int f = lane + 32 * r;
            if (f < F) acc[r] += xr[f] * w;
        }
    }
    float inv = 1.0f / fmaxf((float)(end - beg), 1.0f);
    float* ar = agg + (long)node * F;
#pragma unroll
    for (int r = 0; r < R; r++) {
        int f = lane + 32 * r;
        if (f < F) ar[f] = acc[r] * inv;
    }
}

// ----------------------------------------------- WMMA f32 GEMM + bias + relu
// out[m, 0:256] = relu( [xin_row | agg_row] @ W + bias ),  K = 2F, Kc | K.
// Block = 256 threads (8 waves) computes a 64(M) x 256(N) tile.
// Double-buffered LDS chunks filled with async global->LDS DMA (ASYNCcnt),
// overlapped with WMMA compute on the other buffer.
// W is re-packed as K-pairs Ws[k/2][n][2] (pair-row stride 544 floats) so a
// B fragment {W[k0][n],W[k0+1][n]} is one aligned b64 and lane-half groups
// hit disjoint bank sets; A rows padded to stride Kc+2.
// WMMA f32 16x16x4 layouts:
//   A: lane<16 -> row=lane&15, K-pair {k,k+1}; lane>=16 -> K-pair {k+2,k+3}
//   B: column-major mirror of A; C/D: VGPR r -> row r+8*(lane>=16), col lane&15
template <int F, int Kc>
__global__ void sage_gemm_kernel(const float* __restrict__ xin,
                                 const float* __restrict__ agg,
                                 const float* __restrict__ W,
                                 const float* __restrict__ bias,
                                 float* __restrict__ out, int nn) {
    constexpr int K   = 2 * F;
    constexpr int NCH = K / Kc;         // Kc divides K
    constexpr int AP  = Kc + 2;         // A row stride (floats)
    constexpr int WP  = 2 * HID + 32;   // packed W pair-row stride = 544 floats
    __shared__ float As[2][64 * AP];
    __shared__ float Ws[2][(Kc / 2) * WP];
    const int m0  = blockIdx.x * 64;
    const int tid = threadIdx.x;
    const int wave = tid >> 5;
    const int lane = tid & 31;
    const int half = lane >> 4;
    const int ml   = lane & 15;
    const int mt   = wave >> 1;          // M-subtile 0..3
    const int nb   = (wave & 1) * 128;   // N-half base
    const bool fullM = (m0 + 64 <= nn);

    // async-fill one K-chunk into buffer b (skips invalid M rows)
    auto stage = [&](int b, int kc) {
        for (int idx = tid; idx < 64 * Kc; idx += 256) {
            int r = idx / Kc, c = idx - r * Kc;
            int node = m0 + r;
            if (node < nn) {
                int gk = kc + c;
                const float* g = (gk < F) ? &xin[(long)node * F + gk]
                                          : &agg[(long)node * F + (gk - F)];
                async_copy_b32(lds_off(&As[b][r * AP + c]), g);
            }
        }
        for (int idx = tid; idx < Kc * HID; idx += 256) {
            int kl = idx >> 8, n = idx & (HID - 1);
            async_copy_b32(lds_off(&Ws[b][(kl >> 1) * WP + n * 2 + (kl & 1)]),
                           &W[(long)(kc + kl) * HID + n]);
        }
    };

    v8f acc[8];
#pragma unroll
    for (int j = 0; j < 8; j++) acc[j] = (v8f){};

    stage(0, 0);
    // zero A rows past the valid range once (async stages never touch them)
    if (!fullM) {
        for (int idx = tid; idx < 64 * AP; idx += 256) {
            int r = idx / AP;
            if (m0 + r >= nn) { As[0][idx] = 0.0f; As[1][idx] = 0.0f; }
        }
    }
    wait_async0();
    __syncthreads();

    for (int c = 0; c < NCH; ++c) {
        if (c + 1 < NCH) stage((c + 1) & 1, (c + 1) * Kc);   // fire-and-forget

        const float* arow = &As[c & 1][(mt * 16 + ml) * AP];
        const float* wbuf = &Ws[c & 1][0];
#pragma unroll
        for (int k = 0; k < Kc; k += 4) {
            v2f a = *(const v2f*)&arow[k + half * 2];
            const float* wrow = &wbuf[((k >> 1) + half) * WP];
            v2f b[8];
#pragma unroll
            for (int j = 0; j < 8; j++)
                b[j] = *(const v2f*)&wrow[(nb + j * 16 + ml) * 2];
#pragma unroll
            for (int j = 0; j < 8; j++)
                acc[j] = __builtin_amdgcn_wmma_f32_16x16x4_f32(
                    false, a, false, b[j], (short)0, acc[j], false, false);
        }

        wait_async0();      // my async fills for the next buffer are done
        __syncthreads();    // everyone's fills done; everyone done reading
    }

    // ---- epilogue: bias + relu ----
    const int rowoff = half * 8;
    if (fullM) {
#pragma unroll
        for (int j = 0; j < 8; j++) {
            int n = nb + j * 16 + ml;
            float bv = bias[n];
            float* o = out + (long)(m0 + mt * 16 + rowoff) * HID + n;
#pragma unroll
            for (int r = 0; r < 8; r++)
                o[(long)r * HID] = fmaxf(acc[j][r] + bv, 0.0f);
        }
    } else {
#pragma unroll
        for (int j = 0; j < 8; j++) {
            int n = nb + j * 16 + ml;
            float bv = bias[n];
#pragma unroll
            for (int r = 0; r < 8; r++) {
                int m = m0 + mt * 16 + r + rowoff;
                if (m < nn)
                    out[(long)m * HID + n] = fmaxf(acc[j][r] + bv, 0.0f);
            }
        }
    }
}

// ------------------------------------- final [N,768]@[768,7] + log_softmax
__global__ void final_kernel(const float* __restrict__ x1, const float* __restrict__ x2,
                             const float* __restrict__ x3, const float* __restrict__ Wl,
                             const float* __restrict__ bl, float* __restrict__ out, int n) {
    int node = blockIdx.x * (blockDim.x >> 5) + (threadIdx.x >> 5);
    int lane = threadIdx.x & 31;
    if (node >= n) return;
    float acc[N_CLS];
#pragma unroll
    for (int c = 0; c < N_CLS; c++) acc[c] = 0.0f;
    const float* rows[3] = { x1 + (long)node * HID, x2 + (long)node * HID, x3 + (long)node * HID };
#pragma unroll
    for (int part = 0; part < 3; part++) {
        const float* xr = rows[part];
        for (int k = lane; k < HID; k += 32) {
            float xv = xr[k];
            const float* wr = Wl + (long)(part * HID + k) * N_CLS;
#pragma unroll
            for (int c = 0; c < N_CLS; c++) acc[c] += xv * wr[c];
        }
    }
#pragma unroll
    for (int c = 0; c < N_CLS; c++)
        for (int off = 16; off > 0; off >>= 1)
            acc[c] += __shfl_xor(acc[c], off, 32);
    if (lane == 0) {
        float mx = -1e30f;
#pragma unroll
        for (int c = 0; c < N_CLS; c++) { acc[c] += bl[c]; mx = fmaxf(mx, acc[c]); }
        float s = 0.0f;
#pragma unroll
        for (int c = 0; c < N_CLS; c++) s += expf(acc[c] - mx);
        float lse = mx + logf(s);
        float* o = out + (long)node * N_CLS;
#pragma unroll
        for (int c = 0; c < N_CLS; c++) o[c] = acc[c] - lse;
    }
}

// ---------------------------------------------------------------- launcher
extern "C" void kernel_launch(void* const* d_in, const int* in_sizes, int n_in,
                              void* d_out, int out_size, void* d_ws, size_t ws_size,
                              hipStream_t stream) {
    const float* x  = (const float*)d_in[0];
    const int*   ei = (const int*)d_in[1];
    const float* ew = (const float*)d_in[2];
    const float* W1 = (const float*)d_in[3];
    const float* b1 = (const float*)d_in[4];
    const float* W2 = (const float*)d_in[5];
    const float* b2 = (const float*)d_in[6];
    const float* W3 = (const float*)d_in[7];
    const float* b3 = (const float*)d_in[8];
    const float* Wl = (const float*)d_in[9];
    const float* bl = (const float*)d_in[10];
    float* out = (float*)d_out;

    const int N = in_sizes[0] / IN_F;
    const int E = in_sizes[2];
    const int* src = ei;
    const int* dst = ei + E;

    // workspace carve (256B-aligned slots)
    char* ws = (char*)d_ws;
    size_t off = 0;
    auto carve = [&](size_t bytes) { void* p = ws + off; off = (off + bytes + 255) & ~(size_t)255; return p; };
    int*   ptr  = (int*)carve((size_t)(N + 1) * sizeof(int));
    int*   cnt  = (int*)carve((size_t)N * sizeof(int));        // reused as fill counters
    int*   eid  = (int*)carve((size_t)E * sizeof(int));
    float* agg  = (float*)carve((size_t)N * IN_F * sizeof(float));
    float* x1   = (float*)carve((size_t)N * HID * sizeof(float));
    float* x2   = (float*)carve((size_t)N * HID * sizeof(float));
    float* x3   = (float*)carve((size_t)N * HID * sizeof(float));
    (void)ws_size;

    const int TB = 256;
    dim3 blkN((N + TB - 1) / TB), blkE((E + TB - 1) / TB);
    dim3 waveN((N + 7) / 8);                 // 8 waves (nodes) per 256-thread block
    dim3 gemmN((N + 63) / 64);               // 64-row M-blocks

    // ---- CSR build ----
    zero_int_kernel<<<blkN, TB, 0, stream>>>(cnt, N);
    csr_count_kernel<<<blkE, TB, 0, stream>>>(dst, cnt, E);
    csr_scan_kernel<<<1, 1024, 0, stream>>>(cnt, ptr, N);
    zero_int_kernel<<<blkN, TB, 0, stream>>>(cnt, N);
    csr_fill_kernel<<<blkE, TB, 0, stream>>>(dst, ptr, cnt, eid, E);

    // ---- layer 1: K=1000, Kc=20 (50 chunks) ----
    aggregate_kernel<IN_F><<<waveN, TB, 0, stream>>>(x, ptr, eid, src, ew, agg, N);
    sage_gemm_kernel<IN_F, 20><<<gemmN, TB, 0, stream>>>(x, agg, W1, b1, x1, N);
    // ---- layer 2: K=512, Kc=16 (32 chunks) ----
    aggregate_kernel<HID><<<waveN, TB, 0, stream>>>(x1, ptr, eid, src, ew, agg, N);
    sage_gemm_kernel<HID, 16><<<gemmN, TB, 0, stream>>>(x1, agg, W2, b2, x2, N);
    // ---- layer 3 ----
    aggregate_kernel<HID><<<waveN, TB, 0, stream>>>(x2, ptr, eid, src, ew, agg, N);
    sage_gemm_kernel<HID, 16><<<gemmN, TB, 0, stream>>>(x2, agg, W3, b3, x3, N);
    // ---- classifier + log_softmax ----
    final_kernel<<<waveN, TB, 0, stream>>>(x1, x2, x3, Wl, bl, out, N);
    (void)out_size; (void)n_in;
}